// Discrete_CRPSLoss_29205777613261
// MI455X (gfx1250) — compile-verified
//
#include <hip/hip_runtime.h>
#include <stdint.h>

// ---------------------------------------------------------------------------
// Discrete CRPS loss, CDNA5 (gfx1250).
// forecasts: (50, 64, 336) f32   member-major: fc[j*21504 + cell]
// observation: (64, 336) f32
// out: (64, 336) f32
//
// out(c) = mean_j |clip(x_j)-y| - (1-eps) * sum_{j<k}|x_j-x_k| / (n*(n-1))
// (reference's sum over ordered pairs / (2n(n-1)) == unordered-pair sum / (n(n-1)))
//
// Data movement: one TENSOR_LOAD_TO_LDS per workgroup DMAs the 50x256 f32 tile
// (51200 B) into LDS, tracked with TENSORcnt.
// ---------------------------------------------------------------------------

#define NMEM   50
#define TILE   256
#define NCELLS 21504
#define CLIPV  (-0.26787253f)
#define EPSV   (1e-4f)

typedef __attribute__((ext_vector_type(4))) unsigned int v4u;
typedef __attribute__((ext_vector_type(8))) int          v8i;
typedef __attribute__((ext_vector_type(4))) int          v4i;

__global__ __launch_bounds__(TILE) void
crps_tdm_kernel(const float* __restrict__ fc,
                const float* __restrict__ obs,
                float* __restrict__ out)
{
    __shared__ float tile[NMEM * TILE];      // 51200 bytes, row-major [member][cell]

    const int t     = threadIdx.x;
    const int cell0 = blockIdx.x * TILE;

    // ---- Wave 0 issues the Tensor Data Mover load of the whole tile ----
    if (t < 32) {
        const uint64_t gaddr = (uint64_t)(uintptr_t)(fc + cell0);
        const uint32_t laddr = (uint32_t)(uintptr_t)(&tile[0]); // LDS byte offset

        // D# group 0 (128b): [1:0]=count=1, [63:32]=lds_addr,
        // [120:64]=global_addr, [127:126]=type=2
        v4u g0 = { 1u,
                   laddr,
                   (uint32_t)gaddr,
                   (uint32_t)(gaddr >> 32) | 0x80000000u };

        // D# group 1 (256b):
        //  w0: workgroup_mask=0 | data_size=2 (4B) <<16
        //  w1[31:16] = tensor_dim0[15:0]  (21504)
        //  w2[15:0]  = tensor_dim0[31:16]=0 ; w2[31:16] = tensor_dim1[15:0] (50)
        //  w3[31:16] = tile_dim0 (256)
        //  w4[15:0]  = tile_dim1 (50), tile_dim2=0
        //  w5        = tensor_dim0_stride[31:0] (21504)
        v8i g1 = { (int)0x00020000u,
                   (int)((uint32_t)NCELLS << 16),
                   (int)((uint32_t)NMEM   << 16),
                   (int)((uint32_t)TILE   << 16),
                   (int)NMEM,
                   (int)NCELLS,
                   0, 0 };

        v4i gz = { 0, 0, 0, 0 };
#if __clang_major__ >= 23
        v8i gz8 = { 0, 0, 0, 0, 0, 0, 0, 0 };
        __builtin_amdgcn_tensor_load_to_lds(g0, g1, gz, gz, gz8, 0);
#else
        __builtin_amdgcn_tensor_load_to_lds(g0, g1, gz, gz, 0);
#endif
        __builtin_amdgcn_s_wait_tensorcnt(0);
    }
    __syncthreads();

    // ---- Per-thread CRPS over the 50 ensemble members of one cell ----
    const int   cell = cell0 + t;
    const float y    = obs[cell];

    float x[NMEM];
#pragma unroll
    for (int j = 0; j < NMEM; ++j) {
        // LDS reads: lane-contiguous within a row -> conflict-free ds_load_b32
        x[j] = fmaxf(tile[j * TILE + t], CLIPV);
    }

    float acc1 = 0.0f;                       // sum_j |x_j - y|
#pragma unroll
    for (int j = 0; j < NMEM; ++j)
        acc1 += __builtin_fabsf(x[j] - y);

    float acc2 = 0.0f;                       // sum_{j<k} |x_j - x_k|
#pragma unroll
    for (int j = 0; j < NMEM; ++j) {
#pragma unroll
        for (int k = j + 1; k < NMEM; ++k)
            acc2 += __builtin_fabsf(x[j] - x[k]);
    }

    const float t1 = acc1 * (1.0f / (float)NMEM);
    const float t2 = acc2 * (1.0f / (float)(NMEM * (NMEM - 1)));
    out[cell] = t1 - (1.0f - EPSV) * t2;
}

extern "C" void kernel_launch(void* const* d_in, const int* in_sizes, int n_in,
                              void* d_out, int out_size, void* d_ws, size_t ws_size,
                              hipStream_t stream)
{
    (void)in_sizes; (void)n_in; (void)d_ws; (void)ws_size;
    const float* fc  = (const float*)d_in[0];   // (50, 64, 336) f32
    const float* obs = (const float*)d_in[1];   // (64, 336) f32
    float*       out = (float*)d_out;           // (64, 336) f32

    const int nblocks = (out_size + TILE - 1) / TILE;   // 21504/256 = 84 exact
    crps_tdm_kernel<<<dim3(nblocks), dim3(TILE), 0, stream>>>(fc, obs, out);
}